// ContrastiveHardestNegativeLoss_12713103197212
// MI455X (gfx1250) — compile-verified
//
#include <hip/hip_runtime.h>
#include <hip/hip_bf16.h>
#include <math.h>

typedef __attribute__((ext_vector_type(2))) float v2f;
typedef __attribute__((ext_vector_type(8))) float v8f;

#define POS_THRESH_F 0.1f
#define NEG_THRESH_F 1.4f
#define HASH_SIZE 65536
#define HASH_MASK (HASH_SIZE - 1)
#define EMPTY_KEY 0xFFFFFFFFFFFFFFFFull

__device__ __forceinline__ unsigned int hash_slot(long long key) {
  unsigned long long x = (unsigned long long)key * 0x9E3779B97F4A7C15ull;
  return (unsigned int)(x >> 48) & HASH_MASK;
}

// ---------------------------------------------------------------------------
// Stage 1: gather matched rows, squared norms, per-row positive-loss term
// ---------------------------------------------------------------------------
__global__ __launch_bounds__(256) void k_gather_pos(
    const float* __restrict__ F0, const float* __restrict__ F1,
    const long long* __restrict__ matches,
    float* __restrict__ posF0g, float* __restrict__ posF1g,
    float* __restrict__ nPos0, float* __restrict__ nPos1,
    float* __restrict__ posPart, int P) {
  int wave = threadIdx.x >> 5, lane = threadIdx.x & 31;
  int p = blockIdx.x * 8 + wave;
  __shared__ float sh[8];
  float term = 0.0f;
  if (p < P) {
    long long i0 = matches[2 * p + 0];
    long long i1 = matches[2 * p + 1];
    float f0 = F0[i0 * 32 + lane];
    float f1 = F1[i1 * 32 + lane];
    posF0g[(long long)p * 32 + lane] = f0;
    posF1g[(long long)p * 32 + lane] = f1;
    float s0 = f0 * f0, s1 = f1 * f1;
    float d = f0 - f1;
    float sd = d * d;
#pragma unroll
    for (int off = 16; off >= 1; off >>= 1) {
      s0 += __shfl_xor(s0, off, 32);
      s1 += __shfl_xor(s1, off, 32);
      sd += __shfl_xor(sd, off, 32);
    }
    if (lane == 0) {
      nPos0[p] = s0;
      nPos1[p] = s1;
      term = fmaxf(sd - POS_THRESH_F, 0.0f);
    }
  }
  if (lane == 0) sh[wave] = term;
  __syncthreads();
  if (threadIdx.x == 0) {
    float s = 0.0f;
#pragma unroll
    for (int i = 0; i < 8; ++i) s += sh[i];
    posPart[blockIdx.x] = s;
  }
}

// ---------------------------------------------------------------------------
// Stage 2: gather sel0/sel1 subsets + squared norms
// ---------------------------------------------------------------------------
__global__ __launch_bounds__(256) void k_gather_sub(
    const float* __restrict__ F0, const float* __restrict__ F1,
    const long long* __restrict__ sel0, const long long* __restrict__ sel1,
    float* __restrict__ subF0g, float* __restrict__ subF1g,
    float* __restrict__ nSub0, float* __restrict__ nSub1, int M) {
  int wave = threadIdx.x >> 5, lane = threadIdx.x & 31;
  int m = blockIdx.x * 8 + wave;
  if (m >= M) return;
  long long i0 = sel0[m];
  long long i1 = sel1[m];
  float f0 = F0[i0 * 32 + lane];
  float f1 = F1[i1 * 32 + lane];
  subF0g[(long long)m * 32 + lane] = f0;
  subF1g[(long long)m * 32 + lane] = f1;
  float s0 = f0 * f0, s1 = f1 * f1;
#pragma unroll
  for (int off = 16; off >= 1; off >>= 1) {
    s0 += __shfl_xor(s0, off, 32);
    s1 += __shfl_xor(s1, off, 32);
  }
  if (lane == 0) {
    nSub0[m] = s0;
    nSub1[m] = s1;
  }
}

// ---------------------------------------------------------------------------
// Hash set of pos_keys for the isin() masks
// ---------------------------------------------------------------------------
__global__ void k_hash_clear(unsigned long long* __restrict__ table) {
  int i = blockIdx.x * blockDim.x + threadIdx.x;
  if (i < HASH_SIZE) table[i] = EMPTY_KEY;
}

__global__ void k_hash_insert(const long long* __restrict__ matches,
                              unsigned long long* __restrict__ table,
                              long long seed, int P) {
  int p = blockIdx.x * blockDim.x + threadIdx.x;
  if (p >= P) return;
  long long key = matches[2 * p + 0] + matches[2 * p + 1] * seed;
  unsigned int slot = hash_slot(key);
  for (;;) {
    unsigned long long prev = atomicCAS(&table[slot], EMPTY_KEY, (unsigned long long)key);
    if (prev == EMPTY_KEY || prev == (unsigned long long)key) break;
    slot = (slot + 1) & HASH_MASK;
  }
}

__device__ __forceinline__ bool hash_lookup(const unsigned long long* __restrict__ table,
                                            long long key) {
  unsigned int slot = hash_slot(key);
  for (;;) {
    unsigned long long v = table[slot];
    if (v == EMPTY_KEY) return false;
    if (v == (unsigned long long)key) return true;
    slot = (slot + 1) & HASH_MASK;
  }
}

// ---------------------------------------------------------------------------
// Stage 3: fused GEMM + row min/argmin via V_WMMA_F32_16X16X4_F32.
//
// Each wave owns TWO 16-row tiles of A (32 rows) -> two independent WMMA
// accumulator chains sharing one streamed B fragment: 8 b64 loads feed 16
// WMMAs per column tile, and the two chains interleave so the scheduler
// always has an independent WMMA to issue while the other waits.
//
// Both directions (D01, D10) run in one launch, selected by blockIdx.y,
// to keep 1024 waves in flight.
//
// f32 WMMA fragment layout (ISA 7.12.2): lane l, VGPR v of A(16x4) holds
// A[l&15][2*(l>>4)+v]; B(4x16) mirrors the C/D half-wave split so lane l,
// VGPR v holds B[2*(l>>4)+v][l&15] = subRow[l&15][2*(l>>4)+v] for B = subT.
// => identical float2 gather pattern for both operands.
// ---------------------------------------------------------------------------
__global__ __launch_bounds__(256) void k_hardest_min(
    const float* __restrict__ Ag0, const float* __restrict__ nRow0,
    const float* __restrict__ Bg0, const float* __restrict__ nCol0,
    float* __restrict__ outMin0, int* __restrict__ outCol0,
    const float* __restrict__ Ag1, const float* __restrict__ nRow1,
    const float* __restrict__ Bg1, const float* __restrict__ nCol1,
    float* __restrict__ outMin1, int* __restrict__ outCol1,
    int rowPairs, int Mtiles) {
  const float* Ag   = (blockIdx.y == 0) ? Ag0   : Ag1;
  const float* nRow = (blockIdx.y == 0) ? nRow0 : nRow1;
  const float* Bg   = (blockIdx.y == 0) ? Bg0   : Bg1;
  const float* nCol = (blockIdx.y == 0) ? nCol0 : nCol1;
  float* outMin     = (blockIdx.y == 0) ? outMin0 : outMin1;
  int* outCol       = (blockIdx.y == 0) ? outCol0 : outCol1;

  int lane = threadIdx.x & 31;
  int r = lane & 15;   // column-within-tile / A row
  int h = lane >> 4;   // half-wave select
  int pair = blockIdx.x * 8 + (threadIdx.x >> 5);
  if (pair >= rowPairs) return;
  int R0 = pair * 32;

  const float* abase0 = Ag + ((long long)(R0 + r)) * 32 + 2 * h;
  const float* abase1 = abase0 + 16 * 32;
  v2f a0[8], a1[8];
#pragma unroll
  for (int kc = 0; kc < 8; ++kc) {
    a0[kc] = *(const v2f*)(abase0 + kc * 4);
    a1[kc] = *(const v2f*)(abase1 + kc * 4);
  }

  float runmin0[8], runmin1[8];
  int runcol0[8], runcol1[8];
#pragma unroll
  for (int v = 0; v < 8; ++v) {
    runmin0[v] = 3.4e38f;
    runmin1[v] = 3.4e38f;
    runcol0[v] = 0;
    runcol1[v] = 0;
  }

  for (int ct = 0; ct < Mtiles; ++ct) {
    const float* bbase = Bg + ((long long)(ct * 16 + r)) * 32 + 2 * h;
    // pull the next column tile toward the WGP cache
    __builtin_prefetch(bbase + 16 * 32, 0, 3);
    v2f b[8];
#pragma unroll
    for (int kc = 0; kc < 8; ++kc) b[kc] = *(const v2f*)(bbase + kc * 4);
    float n1v = nCol[ct * 16 + r];

    v8f c0 = {};
    v8f c1 = {};
#pragma unroll
    for (int kc = 0; kc < 8; ++kc) {
      c0 = __builtin_amdgcn_wmma_f32_16x16x4_f32(false, a0[kc], false, b[kc],
                                                 (short)0, c0, false, false);
      c1 = __builtin_amdgcn_wmma_f32_16x16x4_f32(false, a1[kc], false, b[kc],
                                                 (short)0, c1, false, false);
    }

    int col = ct * 16 + r;
#pragma unroll
    for (int v = 0; v < 8; ++v) {
      float val0 = fmaf(-2.0f, c0[v], n1v);  // n1[j] - 2*dot(i,j)
      float val1 = fmaf(-2.0f, c1[v], n1v);
      bool lt0 = val0 < runmin0[v];
      bool lt1 = val1 < runmin1[v];
      runmin0[v] = lt0 ? val0 : runmin0[v];
      runcol0[v] = lt0 ? col : runcol0[v];
      runmin1[v] = lt1 ? val1 : runmin1[v];
      runcol1[v] = lt1 ? col : runcol1[v];
    }
  }

  // reduce across the 16 lanes of each half-wave (row m = v + 8*h)
#pragma unroll
  for (int t = 0; t < 2; ++t) {
#pragma unroll
    for (int v = 0; v < 8; ++v) {
      float m = t == 0 ? runmin0[v] : runmin1[v];
      int cidx = t == 0 ? runcol0[v] : runcol1[v];
#pragma unroll
      for (int off = 8; off >= 1; off >>= 1) {
        float om = __shfl_xor(m, off, 32);
        int oc = __shfl_xor(cidx, off, 32);
        bool take = (om < m) || (om == m && oc < cidx);  // first-occurrence tie-break
        m = take ? om : m;
        cidx = take ? oc : cidx;
      }
      if (r == 0) {
        int row = R0 + t * 16 + v + 8 * h;
        outMin[row] = sqrtf(fmaxf(nRow[row] + m, 0.0f) + 1e-7f);
        outCol[row] = cidx;
      }
    }
  }
}

// ---------------------------------------------------------------------------
// Stage 4: masks via hash probes + final scalar loss (single block)
// ---------------------------------------------------------------------------
__global__ __launch_bounds__(256) void k_final(
    const long long* __restrict__ matches,
    const long long* __restrict__ sel0, const long long* __restrict__ sel1,
    const float* __restrict__ d01min, const int* __restrict__ d01col,
    const float* __restrict__ d10min, const int* __restrict__ d10col,
    const unsigned long long* __restrict__ table,
    const float* __restrict__ posPart, int nPart,
    long long seed, int P, float* __restrict__ out) {
  int tid = threadIdx.x;
  float sp = 0.0f, s0 = 0.0f, s1 = 0.0f, c0 = 0.0f, c1 = 0.0f;
  for (int i = tid; i < nPart; i += 256) sp += posPart[i];
  for (int p = tid; p < P; p += 256) {
    long long i0 = matches[2 * p + 0];
    long long i1 = matches[2 * p + 1];
    long long k0 = i0 + sel1[d01col[p]] * seed;
    if (!hash_lookup(table, k0)) {
      float t = fmaxf(NEG_THRESH_F - d01min[p], 0.0f);
      s0 += t * t;
      c0 += 1.0f;
    }
    long long k1 = sel0[d10col[p]] + i1 * seed;
    if (!hash_lookup(table, k1)) {
      float t = fmaxf(NEG_THRESH_F - d10min[p], 0.0f);
      s1 += t * t;
      c1 += 1.0f;
    }
  }
  __shared__ float sh[256];
  float tot[5];
  float vals[5] = {sp, s0, s1, c0, c1};
#pragma unroll
  for (int k = 0; k < 5; ++k) {
    sh[tid] = vals[k];
    __syncthreads();
    for (int s = 128; s > 0; s >>= 1) {
      if (tid < s) sh[tid] += sh[tid + s];
      __syncthreads();
    }
    tot[k] = sh[0];
    __syncthreads();
  }
  if (tid == 0) {
    float posLoss = tot[0] / (float)P;
    float nl0 = tot[1] / fmaxf(tot[3], 1.0f);
    float nl1 = tot[2] / fmaxf(tot[4], 1.0f);
    out[0] = posLoss + 0.5f * (nl0 + nl1);
  }
}

// ---------------------------------------------------------------------------
extern "C" void kernel_launch(void* const* d_in, const int* in_sizes, int n_in,
                              void* d_out, int out_size, void* d_ws, size_t ws_size,
                              hipStream_t stream) {
  const float* F0 = (const float*)d_in[0];
  const float* F1 = (const float*)d_in[1];
  const long long* matches = (const long long*)d_in[2];
  const long long* sel0 = (const long long*)d_in[3];
  const long long* sel1 = (const long long*)d_in[4];

  const int D = 32;
  const int N0 = in_sizes[0] / D;
  const int N1 = in_sizes[1] / D;
  const int P = in_sizes[2] / 2;
  const int M = in_sizes[3];
  const long long seed = (long long)(N0 > N1 ? N0 : N1);

  // workspace layout
  float* ws = (float*)d_ws;
  float* posF0g = ws;                       // P*32
  float* posF1g = posF0g + (size_t)P * 32;  // P*32
  float* subF0g = posF1g + (size_t)P * 32;  // M*32
  float* subF1g = subF0g + (size_t)M * 32;  // M*32
  float* nPos0 = subF1g + (size_t)M * 32;   // P
  float* nPos1 = nPos0 + P;                 // P
  float* nSub0 = nPos1 + P;                 // M
  float* nSub1 = nSub0 + M;                 // M
  float* d01min = nSub1 + M;                // P
  float* d10min = d01min + P;               // P
  int* d01col = (int*)(d10min + P);         // P
  int* d10col = d01col + P;                 // P
  const int posBlocks = (P + 7) / 8;
  float* posPart = (float*)(d10col + P);    // posBlocks
  size_t off = (size_t)((char*)(posPart + posBlocks) - (char*)d_ws);
  off = (off + 7) & ~(size_t)7;
  unsigned long long* table = (unsigned long long*)((char*)d_ws + off);

  // Stage 1+2: gathers, norms, pos-loss partials
  k_gather_pos<<<posBlocks, 256, 0, stream>>>(F0, F1, matches, posF0g, posF1g,
                                              nPos0, nPos1, posPart, P);
  k_gather_sub<<<(M + 7) / 8, 256, 0, stream>>>(F0, F1, sel0, sel1, subF0g,
                                                subF1g, nSub0, nSub1, M);
  // Hash set of positive keys
  k_hash_clear<<<HASH_SIZE / 256, 256, 0, stream>>>(table);
  k_hash_insert<<<(P + 255) / 256, 256, 0, stream>>>(matches, table, seed, P);

  // Stage 3: fused distance GEMM + hardest-negative min, both directions in
  // one launch (blockIdx.y selects direction)
  const int rowPairs = P / 32;              // 32 rows (two 16-row tiles) / wave
  const int Mtiles = M / 16;
  const int pairBlocks = (rowPairs + 7) / 8;
  dim3 grid(pairBlocks, 2, 1);
  // dir 0 (D01): rows = posF0, cols = subF1 = F1[sel1]
  // dir 1 (D10): rows = posF1, cols = subF0 = F0[sel0]
  k_hardest_min<<<grid, 256, 0, stream>>>(
      posF0g, nPos0, subF1g, nSub1, d01min, d01col,
      posF1g, nPos1, subF0g, nSub0, d10min, d10col,
      rowPairs, Mtiles);

  // Stage 4: masks + final scalar
  k_final<<<1, 256, 0, stream>>>(matches, sel0, sel1, d01min, d01col, d10min,
                                 d10col, table, posPart, posBlocks, seed, P,
                                 (float*)d_out);
}